// AttnBlock_59519656788468
// MI455X (gfx1250) — compile-verified
//
#include <hip/hip_runtime.h>

// AttnBlock collapsed form for MI455X (gfx1250, wave32).
// Sizes fixed by the reference.
#define Bn 16
#define Cc 512
#define HWn 1024
#define Gg 32
#define CPG 16          // channels per group
#define Tt 180
#define Ff 1024
#define EPSf 1e-6f
#define SCALEf 0.044194173824159216f   // 512^-0.5

typedef float v2f __attribute__((ext_vector_type(2)));
typedef float v8f __attribute__((ext_vector_type(8)));

// ---------------- kernel 1: weight pre-reductions -------------------------
// wqc[c] = sum_o wq[o,c]; wsum[o] = sum_c wo[o,c]; bqsum = sum_o bq[o]
__global__ void prep_weights(const float* __restrict__ wq,
                             const float* __restrict__ bq,
                             const float* __restrict__ wo,
                             float* __restrict__ wqc,
                             float* __restrict__ wsum,
                             float* __restrict__ bqsum) {
    int t = threadIdx.x;                 // 512 threads
    float s = 0.f;
    for (int o = 0; o < Cc; ++o) s += wq[o * Cc + t];   // coalesced across t
    wqc[t] = s;
    float s2 = 0.f;
    for (int c = 0; c < Cc; ++c) s2 += wo[t * Cc + c];
    wsum[t] = s2;
    if (t == 0) {
        float s3 = 0.f;
        for (int i = 0; i < Cc; ++i) s3 += bq[i];
        *bqsum = s3;
    }
}

// ---------------- kernel 2: k,v via V_WMMA_F32_16X16X4_F32 ----------------
// k[b,f] = sum_t cond[b,t]*wk[f,t] + bk[f]   (M=16 batches, N=16 f per block,
// K=180 in steps of 4). v likewise, sharing the A operand.
// ISA 7.12.2 32-bit layouts:
//   A 16x4 : lanes 0-15 (M=lane): v0=K0, v1=K1 ; lanes 16-31: v0=K2, v1=K3
//   B 4x16 : N = lane&15 ; v0 holds K0|K2 across lane halves, v1 holds K1|K3
//   C/D    : VGPR r -> M = r + 8*(lane>>4), N = lane&15
__global__ void kv_wmma(const float* __restrict__ cond,
                        const float* __restrict__ wk,
                        const float* __restrict__ bk,
                        const float* __restrict__ wv,
                        const float* __restrict__ bv,
                        float* __restrict__ kbuf,
                        float* __restrict__ vbuf) {
    const int lane = threadIdx.x;        // 32 lanes, EXEC all ones
    const int half = lane >> 4;
    const int mn   = lane & 15;
    const int f0   = blockIdx.x * 16;    // 64 blocks cover F=1024

    v8f ck = {}; v8f cv = {};
    for (int t0 = 0; t0 < Tt; t0 += 4) {
        const int ta = t0 + 2 * half;
        v2f a, bb_k, bb_v;
        a[0] = cond[mn * Tt + ta];
        a[1] = cond[mn * Tt + ta + 1];
        const float* wkr = wk + (size_t)(f0 + mn) * Tt + ta;
        bb_k[0] = wkr[0]; bb_k[1] = wkr[1];
        const float* wvr = wv + (size_t)(f0 + mn) * Tt + ta;
        bb_v[0] = wvr[0]; bb_v[1] = wvr[1];
        ck = __builtin_amdgcn_wmma_f32_16x16x4_f32(false, a, false, bb_k,
                                                   (short)0, ck, false, false);
        cv = __builtin_amdgcn_wmma_f32_16x16x4_f32(false, a, false, bb_v,
                                                   (short)0, cv, false, false);
    }
    const int f = f0 + mn;
    const float bkf = bk[f], bvf = bv[f];
#pragma unroll
    for (int r = 0; r < 8; ++r) {
        const int m = r + 8 * half;      // batch index
        kbuf[m * Ff + f] = ck[r] + bkf;
        vbuf[m * Ff + f] = cv[r] + bvf;
    }
}

// ---------------- kernel 3: GroupNorm statistics --------------------------
__global__ void gn_stats(const float* __restrict__ x,
                         float* __restrict__ mu,
                         float* __restrict__ rstd) {
    const int bg  = blockIdx.x;              // 512 = B*G blocks
    const int tid = threadIdx.x;             // 256 threads
    const float* xp = x + (size_t)bg * (CPG * HWn);   // groups contiguous
    float s = 0.f, ss = 0.f;
    for (int i = tid; i < CPG * HWn; i += 256) {
        float v = xp[i]; s += v; ss += v * v;
    }
    __shared__ float sh0[256], sh1[256];
    sh0[tid] = s; sh1[tid] = ss;
    __syncthreads();
    for (int w = 128; w > 0; w >>= 1) {
        if (tid < w) { sh0[tid] += sh0[tid + w]; sh1[tid] += sh1[tid + w]; }
        __syncthreads();
    }
    if (tid == 0) {
        const float inv = 1.f / (CPG * HWn);
        float m = sh0[0] * inv;
        float v = sh1[0] * inv - m * m;
        mu[bg]   = m;
        rstd[bg] = rsqrtf(v + EPSf);
    }
}

// ---------------- kernel 4: t[b,n] = scale * qs[b,n] ----------------------
// qs[b,n] = sum_c ac[c]*x[b,c,n] + cb + bqsum,
// ac[c] = wqc[c]*gamma[c]*rstd[b,g], cb = sum_c (wqc[c]*beta[c] - ac[c]*mu[b,g])
__global__ void qs_kernel(const float* __restrict__ x,
                          const float* __restrict__ gamma,
                          const float* __restrict__ beta,
                          const float* __restrict__ wqc,
                          const float* __restrict__ bqsum,
                          const float* __restrict__ mu,
                          const float* __restrict__ rstd,
                          float* __restrict__ tbuf) {
    const int b   = blockIdx.x;              // grid (B, 4)
    const int n   = blockIdx.y * 256 + threadIdx.x;
    const int tid = threadIdx.x;
    __shared__ float ac[Cc];
    __shared__ float red[256];
    float cb = 0.f;
    for (int c = tid; c < Cc; c += 256) {
        const int g = c >> 4;                // c / (C/G)
        const float rs = rstd[b * Gg + g];
        const float m  = mu[b * Gg + g];
        const float w  = wqc[c];
        const float a  = w * gamma[c] * rs;
        ac[c] = a;
        cb += w * beta[c] - a * m;
    }
    red[tid] = cb;
    __syncthreads();
    for (int w = 128; w > 0; w >>= 1) {
        if (tid < w) red[tid] += red[tid + w];
        __syncthreads();
    }
    const float cst = red[0] + *bqsum;
    const float* xb = x + (size_t)b * Cc * HWn + n;
    float acc = 0.f;
    for (int c = 0; c < Cc; ++c)             // coalesced across tid
        acc = fmaf(ac[c], xb[(size_t)c * HWn], acc);
    tbuf[b * HWn + n] = (acc + cst) * SCALEf;
}

// ---------------- kernel 5: r[b,n] = sum_f v*softmax(t*k) -----------------
__global__ void attn_r(const float* __restrict__ kbuf,
                       const float* __restrict__ vbuf,
                       const float* __restrict__ tbuf,
                       float* __restrict__ rbuf) {
    const int b   = blockIdx.x;              // grid (B, 4), 256 threads
    const int n   = blockIdx.y * 256 + threadIdx.x;
    const int tid = threadIdx.x;
    __shared__ float kk[Ff], vv[Ff];
    __shared__ float red[256];
    for (int f = tid; f < Ff; f += 256) {
        kk[f] = kbuf[b * Ff + f];
        vv[f] = vbuf[b * Ff + f];
    }
    __syncthreads();
    // block max/min of k[b,:] -> branch-free softmax max: t>=0 ? t*kmax : t*kmin
    float mx = -3.0e38f, mnv = 3.0e38f;
    for (int f = tid; f < Ff; f += 256) {
        mx = fmaxf(mx, kk[f]); mnv = fminf(mnv, kk[f]);
    }
    red[tid] = mx; __syncthreads();
    for (int w = 128; w > 0; w >>= 1) {
        if (tid < w) red[tid] = fmaxf(red[tid], red[tid + w]);
        __syncthreads();
    }
    mx = red[0]; __syncthreads();
    red[tid] = mnv; __syncthreads();
    for (int w = 128; w > 0; w >>= 1) {
        if (tid < w) red[tid] = fminf(red[tid], red[tid + w]);
        __syncthreads();
    }
    mnv = red[0];

    const float t = tbuf[b * HWn + n];
    const float m = (t >= 0.f) ? t * mx : t * mnv;
    float se = 0.f, sve = 0.f;
    for (int f = 0; f < Ff; ++f) {
        const float e = __expf(fmaf(t, kk[f], -m));   // v_exp_f32 path
        se += e;
        sve = fmaf(vv[f], e, sve);
    }
    rbuf[b * HWn + n] = sve / se;
}

// ---------------- kernel 6: out = x + wsum[o]*r[b,n] + bo[o] --------------
__global__ void out_kernel(const float* __restrict__ x,
                           const float* __restrict__ wsum,
                           const float* __restrict__ bo,
                           const float* __restrict__ rbuf,
                           float* __restrict__ out) {
    const size_t i4 = (size_t)blockIdx.x * 256 + threadIdx.x;  // float4 index
    const float4* x4 = (const float4*)x;
    const float4* r4 = (const float4*)rbuf;
    float4* o4 = (float4*)out;
    const int n4 = (int)(i4 & 255);          // 256 float4 per (b,o) row
    const int co = (int)(i4 >> 8);           // b*512 + o
    const int o  = co & 511;
    const int b  = co >> 9;
    const float ws = wsum[o], bb = bo[o];
    const float4 xv = x4[i4];
    const float4 rv = r4[b * 256 + n4];
    float4 ov;
    ov.x = fmaf(ws, rv.x, xv.x) + bb;
    ov.y = fmaf(ws, rv.y, xv.y) + bb;
    ov.z = fmaf(ws, rv.z, xv.z) + bb;
    ov.w = fmaf(ws, rv.w, xv.w) + bb;
    o4[i4] = ov;
}

extern "C" void kernel_launch(void* const* d_in, const int* in_sizes, int n_in,
                              void* d_out, int out_size, void* d_ws, size_t ws_size,
                              hipStream_t stream) {
    const float* x     = (const float*)d_in[0];
    const float* cond  = (const float*)d_in[1];
    const float* gamma = (const float*)d_in[2];
    const float* beta  = (const float*)d_in[3];
    const float* wq    = (const float*)d_in[4];
    const float* bq    = (const float*)d_in[5];
    const float* wk    = (const float*)d_in[6];
    const float* bk    = (const float*)d_in[7];
    const float* wv    = (const float*)d_in[8];
    const float* bv    = (const float*)d_in[9];
    const float* wo    = (const float*)d_in[10];
    const float* bo    = (const float*)d_in[11];
    float* out = (float*)d_out;

    float* ws = (float*)d_ws;                // ~272 KB of f32 scratch
    float* wqc   = ws + 0;                   // 512
    float* wsum  = ws + 512;                 // 512
    float* bqsum = ws + 1024;                // 1
    float* mu    = ws + 2048;                // 512
    float* rstd  = ws + 2560;                // 512
    float* kbuf  = ws + 4096;                // 16384
    float* vbuf  = ws + 20480;               // 16384
    float* tbuf  = ws + 36864;               // 16384
    float* rbuf  = ws + 53248;               // 16384

    prep_weights<<<1, 512, 0, stream>>>(wq, bq, wo, wqc, wsum, bqsum);
    kv_wmma<<<Ff / 16, 32, 0, stream>>>(cond, wk, bk, wv, bv, kbuf, vbuf);
    gn_stats<<<Bn * Gg, 256, 0, stream>>>(x, mu, rstd);
    qs_kernel<<<dim3(Bn, 4), 256, 0, stream>>>(x, gamma, beta, wqc, bqsum,
                                               mu, rstd, tbuf);
    attn_r<<<dim3(Bn, 4), 256, 0, stream>>>(kbuf, vbuf, tbuf, rbuf);
    out_kernel<<<(Bn * Cc * HWn / 4) / 256, 256, 0, stream>>>(x, wsum, bo,
                                                              rbuf, out);
}